// MemoryOperation_63067299774882
// MI455X (gfx1250) — compile-verified
//
#include <hip/hip_runtime.h>
#include <math.h>

#define NN 50000
#define NE 1500000
#define H 128
#define EF 128
#define TD 100
#define KX 484          // msg dim = 128+128+128+100
#define TILE 16
#define KG_IH (KX/4)    // 121 K-groups for W_ih
#define KG_HH (H/4)     // 32  K-groups for W_hh
#define NTILES 24       // 384/16 column tiles
#define NIH (NTILES*KG_IH*64)  // packed W_ih floats = 185856
#define NHH (NTILES*KG_HH*64)  // packed W_hh floats = 49152

typedef float v2f __attribute__((ext_vector_type(2)));
typedef float v8f __attribute__((ext_vector_type(8)));

__device__ __forceinline__ v8f wmma4(v2f a, v2f b, v8f c) {
    return __builtin_amdgcn_wmma_f32_16x16x4_f32(false, a, false, b, (short)0, c,
                                                 false, false);
}

// ---------------- kernel 1: init winner keys ----------------
__global__ void init_win_kernel(unsigned long long* __restrict__ win, int n) {
    int i = blockIdx.x * blockDim.x + threadIdx.x;
    if (i < n) win[i] = 0ull;
}

// ---------------- kernel 2: per-edge argmax(ts, eidx) into dst ----------------
// key = (float_bits(ts) << 32) | (eidx + 1).  ts >= 0 so bits are order-preserving.
// atomicMax picks max ts, then max edge index (reference tiebreak). key==0 => no msg.
__global__ void edge_scan_kernel(const int* __restrict__ dst,
                                 const float* __restrict__ ts,
                                 unsigned long long* __restrict__ win, int e_count) {
    int e = blockIdx.x * blockDim.x + threadIdx.x;
    if (e >= e_count) return;
    unsigned int tbits = __float_as_uint(ts[e]);
    unsigned long long key = ((unsigned long long)tbits << 32) | (unsigned int)(e + 1);
    atomicMax(&win[dst[e]], key);
}

// ---------------- kernel 3: pack W into WMMA B-fragment order ----------------
// Packed[t][kg][lane*2+s] = W[t*16 + (lane&15)][kg*4 + ((lane>>4)<<1) + s]
// so a wave's B-fragment load is one contiguous 256B chunk (coalesced b64/lane).
__global__ void pack_w_kernel(const float* __restrict__ Wih,
                              const float* __restrict__ Whh,
                              float* __restrict__ PWih,
                              float* __restrict__ PWhh) {
    int idx = blockIdx.x * blockDim.x + threadIdx.x;
    if (idx < NIH) {
        int t   = idx / (KG_IH * 64);
        int rem = idx - t * (KG_IH * 64);
        int kg  = rem >> 6;
        int q   = rem & 63;
        int lane = q >> 1, s = q & 1;
        int row = t * 16 + (lane & 15);
        int k   = kg * 4 + ((lane >> 4) << 1) + s;
        PWih[idx] = Wih[(size_t)row * KX + k];
    }
    if (idx < NHH) {
        int t   = idx / (KG_HH * 64);
        int rem = idx - t * (KG_HH * 64);
        int kg  = rem >> 6;
        int q   = rem & 63;
        int lane = q >> 1, s = q & 1;
        int row = t * 16 + (lane & 15);
        int k   = kg * 4 + ((lane >> 4) << 1) + s;
        PWhh[idx] = Whh[(size_t)row * H + k];
    }
}

// ---------------- kernel 4: gather + GRU via f32 WMMA ----------------
// block = 16 nodes, 256 threads = 8 wave32s; wave w owns hidden cols [16w,16w+16).
// B fragments come from the packed (coalesced) weight copies via wave-uniform bases.
__launch_bounds__(256)
__global__ void gru_wmma_kernel(const int* __restrict__ src,
                                const float* __restrict__ efeat,
                                const float* __restrict__ mem,
                                const float* __restrict__ lut,
                                const float* __restrict__ tw,
                                const float* __restrict__ tbias,
                                const float* __restrict__ PWih,  // packed [24][121][64]
                                const float* __restrict__ PWhh,  // packed [24][32][64]
                                const float* __restrict__ bih,   // [384]
                                const float* __restrict__ bhh,   // [384]
                                const unsigned long long* __restrict__ win,
                                float* __restrict__ s_out,
                                float* __restrict__ ts_out) {
    __shared__ float Xs[TILE * KX];      // 16 x 484 message rows (30976 B)
    __shared__ int   m_w[TILE];
    __shared__ int   m_src[TILE];
    __shared__ float m_delta[TILE];
    __shared__ int   m_has[TILE];

    const int tid = threadIdx.x;
    const int i0  = blockIdx.x * TILE;

    // per-row metadata (winning edge, source node, time delta) + ts output
    if (tid < TILE) {
        const int i = i0 + tid;
        const unsigned long long key = win[i];
        const int has = (key != 0ull);
        const int w   = has ? (int)((unsigned int)key) - 1 : 0;
        const int sw  = src[w];
        const float maxts = __uint_as_float((unsigned int)(key >> 32));
        m_w[tid]     = w;
        m_src[tid]   = sw;
        m_has[tid]   = has;
        m_delta[tid] = has ? (maxts - lut[sw]) : 0.0f;
        ts_out[i]    = has ? maxts : lut[i];
    }
    __syncthreads();

    // stage X = [mem[src_w] | mem[i] | efeat[w] | cos(dt*w_t + b_t)] into LDS
    for (int idx = tid; idx < TILE * KX; idx += 256) {
        const int r = idx / KX;
        const int c = idx - r * KX;
        float v;
        if (c < H)               v = mem[(size_t)m_src[r] * H + c];
        else if (c < 2 * H)      v = mem[(size_t)(i0 + r) * H + (c - H)];
        else if (c < 2 * H + EF) v = efeat[(size_t)m_w[r] * EF + (c - 2 * H)];
        else {
            const int t2 = c - (2 * H + EF);
            v = cosf(m_delta[r] * tw[t2] + tbias[t2]);
        }
        Xs[idx] = v;
    }
    __syncthreads();

    const int lane  = tid & 31;
    const int wave  = __builtin_amdgcn_readfirstlane(tid >> 5);  // wave-uniform
    const int lrow  = lane & 15;               // A row / B col within tile
    const int khi   = (lane >> 4) << 1;        // K sub-offset: 0 or 2
    const int lane2 = lane << 1;               // packed-B offset (floats)

    // wave-uniform base + per-lane lane2 -> saddr-form coalesced loads
    const float* Bi0 = PWih + (size_t)((0 * 8 + wave) * KG_IH) * 64 + lane2;
    const float* Bi1 = PWih + (size_t)((1 * 8 + wave) * KG_IH) * 64 + lane2;
    const float* Bi2 = PWih + (size_t)((2 * 8 + wave) * KG_IH) * 64 + lane2;
    const float* Bh0 = PWhh + (size_t)((0 * 8 + wave) * KG_HH) * 64 + lane2;
    const float* Bh1 = PWhh + (size_t)((1 * 8 + wave) * KG_HH) * 64 + lane2;
    const float* Bh2 = PWhh + (size_t)((2 * 8 + wave) * KG_HH) * 64 + lane2;

    const v8f vzero = {0.f, 0.f, 0.f, 0.f, 0.f, 0.f, 0.f, 0.f};
    v8f acc0 = vzero, acc1 = vzero, acc2 = vzero;   // gi_r, gi_z, gi_n
    v8f acc3 = vzero, acc4 = vzero, acc5 = vzero;   // gh_r, gh_z, gh_n

    // ---- gi = X @ W_ih.T : K = 484, 3 gate tiles share the A fragment ----
    const float* Arow = &Xs[lrow * KX];
    for (int kg = 0; kg < KG_IH; ++kg) {
        const int kk = (kg << 2) + khi;
        const v2f a  = *(const v2f*)&Arow[kk];
        const v2f b0 = *(const v2f*)&Bi0[kg * 64];
        const v2f b1 = *(const v2f*)&Bi1[kg * 64];
        const v2f b2 = *(const v2f*)&Bi2[kg * 64];
        acc0 = wmma4(a, b0, acc0);
        acc1 = wmma4(a, b1, acc1);
        acc2 = wmma4(a, b2, acc2);
    }
    // ---- gh = h @ W_hh.T : K = 128, A rows are Xs[.][128:256] (= mem_s) ----
    for (int kg = 0; kg < KG_HH; ++kg) {
        const int kk = H + (kg << 2) + khi;
        const v2f a  = *(const v2f*)&Arow[kk];
        const v2f b0 = *(const v2f*)&Bh0[kg * 64];
        const v2f b1 = *(const v2f*)&Bh1[kg * 64];
        const v2f b2 = *(const v2f*)&Bh2[kg * 64];
        acc3 = wmma4(a, b0, acc3);
        acc4 = wmma4(a, b1, acc4);
        acc5 = wmma4(a, b2, acc5);
    }

    // ---- gates: C layout -> VGPR e holds (row e | e+8), col = lane&15 ----
    const int j = wave * 16 + lrow;            // hidden unit 0..127
    const float b_r  = bih[j]         + bhh[j];
    const float b_z  = bih[H + j]     + bhh[H + j];
    const float bi_n = bih[2 * H + j];
    const float bh_n = bhh[2 * H + j];
    const int mbase  = (lane >> 4) << 3;       // 0 or 8

    #pragma unroll
    for (int e = 0; e < 8; ++e) {
        const int M = e + mbase;
        const float hprev = Xs[M * KX + H + j];
        const float rpre = acc0[e] + acc3[e] + b_r;
        const float zpre = acc1[e] + acc4[e] + b_z;
        const float r = 1.0f / (1.0f + expf(-rpre));
        const float z = 1.0f / (1.0f + expf(-zpre));
        const float n = tanhf(acc2[e] + bi_n + r * (acc5[e] + bh_n));
        const float hnew = (1.0f - z) * n + z * hprev;
        s_out[(size_t)(i0 + M) * H + j] = m_has[M] ? hnew : hprev;
    }
}

extern "C" void kernel_launch(void* const* d_in, const int* in_sizes, int n_in,
                              void* d_out, int out_size, void* d_ws, size_t ws_size,
                              hipStream_t stream) {
    const int*   src   = (const int*)d_in[0];
    const int*   dst   = (const int*)d_in[1];
    const float* e_ts  = (const float*)d_in[2];
    const float* efeat = (const float*)d_in[3];
    const float* mem   = (const float*)d_in[4];
    const float* lut   = (const float*)d_in[5];
    const float* tw    = (const float*)d_in[6];
    const float* tb    = (const float*)d_in[7];
    const float* Wih   = (const float*)d_in[8];
    const float* Whh   = (const float*)d_in[9];
    const float* bih   = (const float*)d_in[10];
    const float* bhh   = (const float*)d_in[11];

    float* s_out  = (float*)d_out;                        // [N,128]
    float* ts_out = s_out + (size_t)NN * H;               // [N]

    // workspace layout: [0, 400000) winner keys; then packed weights
    unsigned long long* win = (unsigned long long*)d_ws;  // NN * 8 B
    float* PWih = (float*)((char*)d_ws + 409600);         // NIH floats
    float* PWhh = PWih + NIH;                             // NHH floats

    init_win_kernel<<<(NN + 255) / 256, 256, 0, stream>>>(win, NN);
    edge_scan_kernel<<<(NE + 255) / 256, 256, 0, stream>>>(dst, e_ts, win, NE);
    pack_w_kernel<<<(NIH + 255) / 256, 256, 0, stream>>>(Wih, Whh, PWih, PWhh);
    gru_wmma_kernel<<<NN / TILE, 256, 0, stream>>>(src, efeat, mem, lut, tw, tb,
                                                   PWih, PWhh, bih, bhh, win,
                                                   s_out, ts_out);
}